// StaticGNN_34531537060254
// MI455X (gfx1250) — compile-verified
//
#include <hip/hip_runtime.h>

#define N_NODES 50000
#define N_EDGES 600000
#define DIM     128
#define NGRAPH  32
#define DOUT    3

typedef __attribute__((ext_vector_type(2))) float v2f;
typedef __attribute__((ext_vector_type(8))) float v8f;

// ---------------------------------------------------------------- utilities
__global__ void fill_f32(float* __restrict__ p, float v, int n) {
    int i = blockIdx.x * blockDim.x + threadIdx.x;
    if (i < n) p[i] = v;
}

// in-degree counts (float) via global atomics
__global__ void degree_kernel(const int* __restrict__ dst, float* __restrict__ cnt, int ne) {
    int e = blockIdx.x * blockDim.x + threadIdx.x;
    if (e < ne) atomicAdd(&cnt[dst[e]], 1.0f);
}

// cnt -> 1/max(cnt,1)
__global__ void recip_kernel(float* __restrict__ cnt, int n) {
    int i = blockIdx.x * blockDim.x + threadIdx.x;
    if (i < n) cnt[i] = 1.0f / fmaxf(cnt[i], 1.0f);
}

// per-graph node counts
__global__ void gcount_kernel(const int* __restrict__ batch, float* __restrict__ gcnt, int n) {
    int i = blockIdx.x * blockDim.x + threadIdx.x;
    if (i < n) atomicAdd(&gcnt[batch[i]], 1.0f);
}

// one wave per edge: lane does a float4 slice of the 128-float feature row.
// reads are fully coalesced (wave covers exactly one 512B row); adds hit L2
// (ssum = 25.6MB, fully L2-resident on the 192MB L2).
__global__ __launch_bounds__(256) void scatter_kernel(
    const float* __restrict__ feat, const int* __restrict__ src,
    const int* __restrict__ dst, float* __restrict__ ssum, int ne) {
    int wave = blockIdx.x * (blockDim.x >> 5) + (threadIdx.x >> 5);
    int lane = threadIdx.x & 31;
    if (wave >= ne) return;
    int s = src[wave];
    int d = dst[wave];
    const float4 v = *(const float4*)(feat + (size_t)s * DIM + lane * 4);
    float* o = ssum + (size_t)d * DIM + lane * 4;
    atomicAdd(o + 0, v.x);
    atomicAdd(o + 1, v.y);
    atomicAdd(o + 2, v.z);
    atomicAdd(o + 3, v.w);
}

// ---------------------------------------------------------------- WMMA GEMM
// Computes  out = relu( (A0 @ W0^T) * rowscale  [+ A1 @ W1^T]  + bias )
// (rowscale distributes over the K-sum, so it is applied in the epilogue.)
// A0/A1: [N,128] row-major, W: [128out,128in] row-major (B-tile = W^T tile).
// One wave -> one 16x16 output tile, K chained via V_WMMA_F32_16X16X4_F32.
// TWO independent accumulator chains (acc0/acc1) so the scheduler can
// pipeline the XDL WMMA latency instead of serializing on one RAW chain.
// If out==nullptr: fused global-mean-pool epilogue (atomicAdd into gsum[batch]).
__global__ __launch_bounds__(256) void sage_wmma_kernel(
    const float* __restrict__ A0, const float* __restrict__ rowscale,
    const float* __restrict__ W0,
    const float* __restrict__ A1, const float* __restrict__ W1,
    const float* __restrict__ bias,
    float* __restrict__ out,
    const int* __restrict__ batch, float* __restrict__ gsum,
    int mtiles) {
    int wave = threadIdx.x >> 5;
    int lane = threadIdx.x & 31;
    int tile = blockIdx.x * (blockDim.x >> 5) + wave; // 8 waves/block
    int mtile = tile >> 3;                            // o-tile fastest: block shares A rows
    int otile = tile & 7;
    if (mtile >= mtiles) return;                      // wave-uniform; EXEC stays all-1s

    int m0   = mtile * 16;
    int o0   = otile * 16;
    int half = lane >> 4;      // selects K {0,1} vs {2,3} within a K=4 step
    int l15  = lane & 15;
    int arow = m0 + l15;       // A-matrix row this lane supplies (M = lane%16)
    int wrow = o0 + l15;       // W row (output feature) this lane supplies for B

    v8f acc0 = {0.f, 0.f, 0.f, 0.f, 0.f, 0.f, 0.f, 0.f};
    v8f acc1 = {0.f, 0.f, 0.f, 0.f, 0.f, 0.f, 0.f, 0.f};

    const float* a0p = A0 + (size_t)arow * DIM + half * 2;
    const float* w0p = W0 + (size_t)wrow * DIM + half * 2;

    if (A1) {
        // dual GEMM: interleave the two independent accumulation chains
        const float* a1p = A1 + (size_t)arow * DIM + half * 2;
        const float* w1p = W1 + (size_t)wrow * DIM + half * 2;
#pragma unroll
        for (int k = 0; k < DIM; k += 4) {
            v2f a0 = *(const v2f*)(a0p + k);  // VGPR0 = K even, VGPR1 = K odd
            v2f b0 = *(const v2f*)(w0p + k);
            acc0 = __builtin_amdgcn_wmma_f32_16x16x4_f32(
                false, a0, false, b0, (short)0, acc0, false, false);
            v2f a1 = *(const v2f*)(a1p + k);
            v2f b1 = *(const v2f*)(w1p + k);
            acc1 = __builtin_amdgcn_wmma_f32_16x16x4_f32(
                false, a1, false, b1, (short)0, acc1, false, false);
        }
    } else {
        // single GEMM: split K over two chains to halve the RAW depth
#pragma unroll
        for (int k = 0; k < DIM; k += 8) {
            v2f a0 = *(const v2f*)(a0p + k);
            v2f b0 = *(const v2f*)(w0p + k);
            acc0 = __builtin_amdgcn_wmma_f32_16x16x4_f32(
                false, a0, false, b0, (short)0, acc0, false, false);
            v2f a1 = *(const v2f*)(a0p + k + 4);
            v2f b1 = *(const v2f*)(w0p + k + 4);
            acc1 = __builtin_amdgcn_wmma_f32_16x16x4_f32(
                false, a1, false, b1, (short)0, acc1, false, false);
        }
    }

    // C/D layout: VGPR r = row m0 + r + half*8, column = o0 + lane%16
    int col = o0 + l15;
    float bv = bias[col];
    int rbase = m0 + half * 8;
    if (out) {
#pragma unroll
        for (int r = 0; r < 8; ++r) {
            int row = rbase + r;
            float g0 = acc0[r];
            if (rowscale) g0 *= rowscale[row];    // deferred mean-aggregation divide
            float v = g0 + acc1[r] + bv;
            out[(size_t)row * DIM + col] = v > 0.f ? v : 0.f;
        }
    } else {
#pragma unroll
        for (int r = 0; r < 8; ++r) {
            int row = rbase + r;
            float g0 = acc0[r];
            if (rowscale) g0 *= rowscale[row];
            float v = g0 + acc1[r] + bv;
            v = v > 0.f ? v : 0.f;
            atomicAdd(&gsum[batch[row] * DIM + col], v); // fused mean-pool
        }
    }
}

// ---------------------------------------------------------------- final FC
// out[g,o] = (gsum[g,:] . fcW[o,:]) / gcnt[g] + fcb[o]   (32x3 = 96 outputs)
__global__ void fc_kernel(const float* __restrict__ gsum, const float* __restrict__ gcnt,
                          const float* __restrict__ fcW, const float* __restrict__ fcb,
                          float* __restrict__ outp) {
    int t = blockIdx.x * blockDim.x + threadIdx.x;
    if (t >= NGRAPH * DOUT) return;
    int g = t / DOUT, o = t % DOUT;
    float inv = 1.0f / fmaxf(gcnt[g], 1.0f);
    float s = 0.f;
#pragma unroll 4
    for (int k = 0; k < DIM; ++k) s += gsum[g * DIM + k] * fcW[o * DIM + k];
    outp[t] = s * inv + fcb[o];
}

// ---------------------------------------------------------------- launcher
extern "C" void kernel_launch(void* const* d_in, const int* in_sizes, int n_in,
                              void* d_out, int out_size, void* d_ws, size_t ws_size,
                              hipStream_t stream) {
    const float* x   = (const float*)d_in[0];
    const float* Wl0 = (const float*)d_in[1];
    const float* bl0 = (const float*)d_in[2];
    const float* Wr0 = (const float*)d_in[3];
    const float* W1  = (const float*)d_in[4];
    const float* b1  = (const float*)d_in[5];
    const float* Wl1 = (const float*)d_in[6];
    const float* bl1 = (const float*)d_in[7];
    const float* Wr1 = (const float*)d_in[8];
    const float* fcW = (const float*)d_in[9];
    const float* fcb = (const float*)d_in[10];
    const int* edge  = (const int*)d_in[11];   // [2,E] int32 (JAX default x64=off)
    const int* batch = (const int*)d_in[12];
    const int* src = edge;
    const int* dst = edge + N_EDGES;

    // workspace layout (floats); total ~77 MB
    float* ws   = (float*)d_ws;
    float* invc = ws;                              // N   (degree, then 1/max(deg,1))
    float* ssum = ws + 50048;                      // N*128 neighbor-sum scratch
    float* h0   = ssum + (size_t)N_NODES * DIM;    // N*128 conv0 output
    float* h1   = h0   + (size_t)N_NODES * DIM;    // N*128 lin1 output
    float* gsum = h1   + (size_t)N_NODES * DIM;    // 32*128 pooled sums
    float* gcnt = gsum + NGRAPH * DIM;             // 32

    const int NF = N_NODES * DIM;

    // zero scratch (harness does not re-zero between replays)
    fill_f32<<<(50048 + 255) / 256, 256, 0, stream>>>(invc, 0.f, 50048);
    fill_f32<<<(NF + 255) / 256, 256, 0, stream>>>(ssum, 0.f, NF);
    fill_f32<<<(NGRAPH * DIM + NGRAPH + 255) / 256, 256, 0, stream>>>(gsum, 0.f, NGRAPH * DIM + NGRAPH);

    // degrees and graph sizes
    degree_kernel<<<(N_EDGES + 255) / 256, 256, 0, stream>>>(dst, invc, N_EDGES);
    recip_kernel<<<(N_NODES + 255) / 256, 256, 0, stream>>>(invc, N_NODES);
    gcount_kernel<<<(N_NODES + 255) / 256, 256, 0, stream>>>(batch, gcnt, N_NODES);

    const int mtiles   = N_NODES / 16;        // 3125 (exact)
    const int gemmGrid = mtiles;              // 8 waves/block * 8 o-tiles
    const int scatGrid = N_EDGES / 8;         // 8 edges (waves) per 256-thr block

    // conv0: h0 = relu( mean-aggr(x) @ Wl0^T + bl0 + x @ Wr0^T )
    scatter_kernel<<<scatGrid, 256, 0, stream>>>(x, src, dst, ssum, N_EDGES);
    sage_wmma_kernel<<<gemmGrid, 256, 0, stream>>>(ssum, invc, Wl0, x, Wr0, bl0,
                                                   h0, nullptr, nullptr, mtiles);

    // lin1: h1 = relu( h0 @ W1^T + b1 )
    sage_wmma_kernel<<<gemmGrid, 256, 0, stream>>>(h0, nullptr, W1, nullptr, nullptr, b1,
                                                   h1, nullptr, nullptr, mtiles);

    // conv1 (+ fused global mean-pool sums): gsum += relu(sage1(h1))
    fill_f32<<<(NF + 255) / 256, 256, 0, stream>>>(ssum, 0.f, NF);
    scatter_kernel<<<scatGrid, 256, 0, stream>>>(h1, src, dst, ssum, N_EDGES);
    sage_wmma_kernel<<<gemmGrid, 256, 0, stream>>>(ssum, invc, Wl1, h1, Wr1, bl1,
                                                   nullptr, batch, gsum, mtiles);

    // final FC on pooled means -> [32,3]
    fc_kernel<<<1, 128, 0, stream>>>(gsum, gcnt, fcW, fcb, (float*)d_out);
}